// DynamicGNN_84988812853883
// MI455X (gfx1250) — compile-verified
//
#include <hip/hip_runtime.h>
#include <hip/hip_bf16.h>

// ---------------------------------------------------------------------------
// Two-layer GCN forward for MI455X (gfx1250).
// Roofline: scatter/gather traffic ~2 GB -> ~90us @ 23.3 TB/s dominates;
// GEMMs are 2.5 GFLOP total -> negligible. Dense transforms use fp32-exact
// V_WMMA_F32_16X16X4_F32 (precision parity with fp32 reference). Each wave
// computes a full 16xNOUT strip so every A fragment is loaded exactly once
// (X read 1x instead of 8x). Edge aggregation = wave-per-edge coalesced
// b128 gathers + global_atomic_add_f32 scatter.
// ---------------------------------------------------------------------------

#define GCN_N_NODES 50000
#define GCN_N_EDGES 800000
#define GCN_IN_DIM  128
#define GCN_HID_DIM 128
#define GCN_OUT_DIM 64

typedef __attribute__((ext_vector_type(2))) float v2f;
typedef __attribute__((ext_vector_type(8))) float v8f;

// ---- degree / normalization -----------------------------------------------

__global__ void gcn_init_deg(float* __restrict__ deg, int n) {
    int i = blockIdx.x * blockDim.x + threadIdx.x;
    if (i < n) deg[i] = 1.0f;  // self-loop contributes 1 to every node
}

__global__ void gcn_deg_accum(const int* __restrict__ dst, float* __restrict__ deg, int e) {
    int i = blockIdx.x * blockDim.x + threadIdx.x;
    if (i < e) atomicAdd(&deg[dst[i]], 1.0f);
}

__global__ void gcn_finalize_dinv(float* __restrict__ deg, int n) {
    int i = blockIdx.x * blockDim.x + threadIdx.x;
    if (i < n) deg[i] = rsqrtf(fmaxf(deg[i], 1.0f));  // deg -> dinv in place
}

// ---- dense transform: C[M,NOUT] = (RELU? relu(A):A)[M,K] @ W[K,NOUT] -------
// One wave per 16-row strip, computing all NOUT/16 tiles so the A fragment is
// loaded once per K-step and reused across 8 (resp. 4) consecutive WMMAs.
// A-fragment (16x4 f32, 2 VGPRs/lane): row m = lane%16, K = kbase+2*(lane/16)+j
// C/D (8 VGPRs): element r <-> M = r + 8*(lane/16), N = lane%16 (ISA 7.12.2).
template <int K, int NOUT, bool RELU>
__global__ __launch_bounds__(32) void gcn_gemm_wmma(const float* __restrict__ A,
                                                    const float* __restrict__ W,
                                                    float* __restrict__ C) {
    constexpr int NT = NOUT / 16;        // N-tiles per wave (8 or 4)
    const int lane  = threadIdx.x & 31;
    const int tileM = blockIdx.x;        // 50000/16 = 3125, exact
    const int mrow  = lane & 15;
    const int khalf = lane >> 4;         // 0 or 1

    const float* arow  = A + (size_t)(tileM * 16 + mrow) * K;
    const float* wbase = W + mrow;       // N index within tile == lane%16

    v8f acc[NT];
#pragma unroll
    for (int t = 0; t < NT; ++t) acc[t] = (v8f){};

#pragma unroll 2
    for (int k0 = 0; k0 < K; k0 += 4) {
        const int ka = k0 + 2 * khalf;   // even -> 8B-aligned b64 load
        v2f a = *(const v2f*)(arow + ka);
        if (RELU) {
            a.x = fmaxf(a.x, 0.0f);
            a.y = fmaxf(a.y, 0.0f);
        }
#pragma unroll
        for (int t = 0; t < NT; ++t) {
            v2f b;
            b.x = wbase[(size_t)ka * NOUT + t * 16];
            b.y = wbase[(size_t)(ka + 1) * NOUT + t * 16];
            acc[t] = __builtin_amdgcn_wmma_f32_16x16x4_f32(
                /*neg_a=*/false, a, /*neg_b=*/false, b,
                /*c_mod=*/(short)0, acc[t], /*reuse_a=*/false, /*reuse_b=*/false);
        }
    }

    float* cbase = C + (size_t)(tileM * 16 + khalf * 8) * NOUT + mrow;
#pragma unroll
    for (int t = 0; t < NT; ++t) {
#pragma unroll
        for (int r = 0; r < 8; ++r) cbase[(size_t)r * NOUT + t * 16] = acc[t][r];
    }
}

// ---- self-loop + bias init: out[i][f] = b[f] + h[i][f] * dinv[i]^2 ---------
// Vectorized float4: each thread owns 4 consecutive features of one node.
template <int F>
__global__ void gcn_selfloop_bias(const float* __restrict__ h,
                                  const float* __restrict__ dinv,
                                  const float* __restrict__ bias,
                                  float* __restrict__ out, int n) {
    int t = blockIdx.x * blockDim.x + threadIdx.x;   // quad index
    if (t < n * (F / 4)) {
        int i  = t / (F / 4);
        int f4 = t - i * (F / 4);
        float di = dinv[i];
        float s  = di * di;
        float4 hv = ((const float4*)h)[t];
        float4 bv = ((const float4*)bias)[f4];
        float4 o;
        o.x = bv.x + hv.x * s;
        o.y = bv.y + hv.y * s;
        o.z = bv.z + hv.z * s;
        o.w = bv.w + hv.w * s;
        ((float4*)out)[t] = o;
    }
}

// ---- edge scatter: out[dst[e]] += h[src[e]] * dinv[src]*dinv[dst] ----------
// One wave per edge; lane owns a contiguous F/32-float chunk -> b128/b64
// gathers, then scalar global_atomic_add_f32 scatter (no packed f32 atomic).
template <int F>
__global__ void gcn_edge_scatter(const float* __restrict__ h,
                                 const float* __restrict__ dinv,
                                 const int* __restrict__ src,
                                 const int* __restrict__ dst,
                                 float* __restrict__ out, int nEdges) {
    constexpr int VEC = F / 32;  // 4 (F=128) or 2 (F=64)
    const int wavesPerBlock = blockDim.x >> 5;
    const int wave   = blockIdx.x * wavesPerBlock + (threadIdx.x >> 5);
    const int lane   = threadIdx.x & 31;
    const int nWaves = gridDim.x * wavesPerBlock;
    for (int e = wave; e < nEdges; e += nWaves) {
        const int s = src[e];
        const int d = dst[e];
        const float nrm = dinv[s] * dinv[d];
        const float* hs = h + (size_t)s * F + lane * VEC;
        float* od = out + (size_t)d * F + lane * VEC;
        if constexpr (VEC == 4) {
            float4 v = *(const float4*)hs;          // global_load_b128
            atomicAdd(&od[0], v.x * nrm);
            atomicAdd(&od[1], v.y * nrm);
            atomicAdd(&od[2], v.z * nrm);
            atomicAdd(&od[3], v.w * nrm);
        } else {
            float2 v = *(const float2*)hs;          // global_load_b64
            atomicAdd(&od[0], v.x * nrm);
            atomicAdd(&od[1], v.y * nrm);
        }
    }
}

// ---------------------------------------------------------------------------

extern "C" void kernel_launch(void* const* d_in, const int* in_sizes, int n_in,
                              void* d_out, int out_size, void* d_ws, size_t ws_size,
                              hipStream_t stream) {
    const float* x  = (const float*)d_in[0];
    const int*   ei = (const int*)d_in[1];   // [2, E] flat: row0 = src, row1 = dst
    const float* W1 = (const float*)d_in[2];
    const float* b1 = (const float*)d_in[3];
    const float* W2 = (const float*)d_in[4];
    const float* b2 = (const float*)d_in[5];
    float* out = (float*)d_out;

    const int* src = ei;
    const int* dst = ei + GCN_N_EDGES;

    // Workspace carve-up (floats): dinv | h1 | agg1 | h2   (~64.2 MB total)
    float* dinv = (float*)d_ws;
    float* h1   = dinv + GCN_N_NODES;                        // N*128
    float* agg1 = h1   + (size_t)GCN_N_NODES * GCN_HID_DIM;  // N*128
    float* h2   = agg1 + (size_t)GCN_N_NODES * GCN_HID_DIM;  // N*64

    const int nodeBlocks = (GCN_N_NODES + 255) / 256;
    const int edgeBlocks = (GCN_N_EDGES + 255) / 256;

    // 1) symmetric-normalization coefficients
    gcn_init_deg<<<nodeBlocks, 256, 0, stream>>>(dinv, GCN_N_NODES);
    gcn_deg_accum<<<edgeBlocks, 256, 0, stream>>>(dst, dinv, GCN_N_EDGES);
    gcn_finalize_dinv<<<nodeBlocks, 256, 0, stream>>>(dinv, GCN_N_NODES);

    // 2) layer 1 dense: h1 = x @ W1  (3125 waves, each a 16x128 strip)
    gcn_gemm_wmma<GCN_IN_DIM, GCN_HID_DIM, false>
        <<<GCN_N_NODES / 16, 32, 0, stream>>>(x, W1, h1);

    // 3) layer 1 aggregate: agg1 = b1 + selfloop + edge scatter
    gcn_selfloop_bias<GCN_HID_DIM>
        <<<(GCN_N_NODES * (GCN_HID_DIM / 4) + 255) / 256, 256, 0, stream>>>(
            h1, dinv, b1, agg1, GCN_N_NODES);
    gcn_edge_scatter<GCN_HID_DIM>
        <<<4096, 256, 0, stream>>>(h1, dinv, src, dst, agg1, GCN_N_EDGES);

    // 4) layer 2 dense (ReLU fused into A-fragment load): h2 = relu(agg1) @ W2
    gcn_gemm_wmma<GCN_HID_DIM, GCN_OUT_DIM, true>
        <<<GCN_N_NODES / 16, 32, 0, stream>>>(agg1, W2, h2);

    // 5) layer 2 aggregate straight into d_out
    gcn_selfloop_bias<GCN_OUT_DIM>
        <<<(GCN_N_NODES * (GCN_OUT_DIM / 4) + 255) / 256, 256, 0, stream>>>(
            h2, dinv, b2, out, GCN_N_NODES);
    gcn_edge_scatter<GCN_OUT_DIM>
        <<<4096, 256, 0, stream>>>(h2, dinv, src, dst, out, GCN_N_EDGES);
}